// SpatialBlock_43035572306760
// MI455X (gfx1250) — compile-verified
//
#include <hip/hip_runtime.h>

// GCNConv: B=4, N=50000, E=800000, C_IN=32, C_OUT=64
// out[b,d,:] = bias + sum_{edges e: dst=d} norm[e]*h[b,src[e],:] + dinv[d]^2 * h[b,d,:]
// h = x @ W, norm[e] = dinv[src]*w*dinv[dst], dinv = rsqrt(1 + sum_{dst} w)

#define C_IN   32
#define C_OUT  64
#define BATCH  4
#define LDS_STRIDE (C_OUT + 16)   // 80: half-wave rows land in disjoint LDS bank ranges

typedef __attribute__((ext_vector_type(2))) float v2f;
typedef __attribute__((ext_vector_type(8))) float v8f;

// ---------------------------------------------------------------------------
// deg/dinv/norm helpers
// ---------------------------------------------------------------------------
__global__ void deg_init_kernel(float* __restrict__ deg, int N) {
  int i = blockIdx.x * blockDim.x + threadIdx.x;
  if (i < N) deg[i] = 1.0f;                 // self-loop weight
}

__global__ void deg_accum_kernel(const int* __restrict__ dst,
                                 const float* __restrict__ w,
                                 float* __restrict__ deg, int E) {
  int e = blockIdx.x * blockDim.x + threadIdx.x;
  if (e < E) atomicAdd(&deg[dst[e]], w[e]);
}

__global__ void deg_to_dinv_kernel(float* __restrict__ deg, int N) {
  int i = blockIdx.x * blockDim.x + threadIdx.x;
  if (i < N) deg[i] = rsqrtf(deg[i]);       // deg >= 1 always (self-loop)
}

__global__ void edge_norm_kernel(const int* __restrict__ src,
                                 const int* __restrict__ dst,
                                 const float* __restrict__ w,
                                 const float* __restrict__ dinv,
                                 float* __restrict__ norm, int E) {
  int e = blockIdx.x * blockDim.x + threadIdx.x;
  if (e < E) norm[e] = dinv[src[e]] * w[e] * dinv[dst[e]];
}

// ---------------------------------------------------------------------------
// h[r, 0:64] = x[r, 0:32] @ W[32,64], r = b*N + n  (rows = B*N, multiple of 16)
// One wave per 16-row tile; 4 column tiles of 16; K=32 as 8 chained 16x16x4
// f32 WMMAs per column tile.
// ---------------------------------------------------------------------------
__global__ __launch_bounds__(256) void gemm_wmma_kernel(
    const float* __restrict__ x, const float* __restrict__ W,
    float* __restrict__ h, int rows)
{
  __shared__ float lW[C_IN * LDS_STRIDE];
  // Cooperative preload of W (32x64) into padded LDS, before any wave exits.
  for (int i = threadIdx.x; i < C_IN * C_OUT; i += 256) {
    int k = i >> 6, n = i & 63;
    lW[k * LDS_STRIDE + n] = W[i];
  }
  __syncthreads();

  const int wave = threadIdx.x >> 5;
  const int lane = threadIdx.x & 31;
  const int tile = blockIdx.x * 8 + wave;
  if (tile * 16 >= rows) return;            // wave-uniform: EXEC stays all-ones
  const int row0  = tile * 16;
  const int col   = lane & 15;
  const int khalf = (lane >> 4) << 1;       // 0 for lanes 0-15, 2 for lanes 16-31

  v8f acc0 = {}, acc1 = {}, acc2 = {}, acc3 = {};
#pragma unroll
  for (int k0 = 0; k0 < C_IN; k0 += 4) {
    // A (16x4 f32): lane holds (m=col, K=k0+khalf) and (m=col, K=k0+khalf+1)
    v2f a = *(const v2f*)(x + (size_t)(row0 + col) * C_IN + k0 + khalf);
    // B (4x16 f32): VGPR j = row k0+khalf+j for this half-wave, column `col`
    const float* w0 = &lW[(k0 + khalf) * LDS_STRIDE + col];
    const float* w1 = w0 + LDS_STRIDE;
    v2f b0; b0.x = w0[ 0]; b0.y = w1[ 0];
    v2f b1; b1.x = w0[16]; b1.y = w1[16];
    v2f b2; b2.x = w0[32]; b2.y = w1[32];
    v2f b3; b3.x = w0[48]; b3.y = w1[48];
    acc0 = __builtin_amdgcn_wmma_f32_16x16x4_f32(false, a, false, b0, (short)0, acc0, false, false);
    acc1 = __builtin_amdgcn_wmma_f32_16x16x4_f32(false, a, false, b1, (short)0, acc1, false, false);
    acc2 = __builtin_amdgcn_wmma_f32_16x16x4_f32(false, a, false, b2, (short)0, acc2, false, false);
    acc3 = __builtin_amdgcn_wmma_f32_16x16x4_f32(false, a, false, b3, (short)0, acc3, false, false);
  }

  // C/D 16x16 f32: VGPR i -> row i (lanes 0-15) / row i+8 (lanes 16-31)
  const int mrow = (lane >> 4) << 3;
  float* hp = h + (size_t)(row0 + mrow) * C_OUT + col;
#pragma unroll
  for (int i = 0; i < 8; ++i) {
    hp[(size_t)i * C_OUT +  0] = acc0[i];
    hp[(size_t)i * C_OUT + 16] = acc1[i];
    hp[(size_t)i * C_OUT + 32] = acc2[i];
    hp[(size_t)i * C_OUT + 48] = acc3[i];
  }
}

// ---------------------------------------------------------------------------
// out[b,n,c] = h[b,n,c] * dinv[n]^2 + bias[c]   (self-loop message + bias,
// also serves as the zero-base initializer for the atomic scatter)
// One thread per float4; total4 = B*N*16
// ---------------------------------------------------------------------------
__global__ void out_init_kernel(const float* __restrict__ h,
                                const float* __restrict__ dinv,
                                const float* __restrict__ bias,
                                float* __restrict__ out, int N, size_t total4) {
  size_t t = (size_t)blockIdx.x * blockDim.x + threadIdx.x;
  if (t >= total4) return;
  int c4      = (int)(t & 15);
  size_t bn   = t >> 4;
  int node    = (int)(bn % (size_t)N);
  float s     = dinv[node];
  float ss    = s * s;
  float4 hv   = ((const float4*)h)[t];
  float4 bb   = ((const float4*)bias)[c4];
  float4 o;
  o.x = hv.x * ss + bb.x;
  o.y = hv.y * ss + bb.y;
  o.z = hv.z * ss + bb.z;
  o.w = hv.w * ss + bb.w;
  ((float4*)out)[t] = o;
}

// ---------------------------------------------------------------------------
// Edge scatter: thread -> (e, b, c4). float4 gather of h[b,src], 4 f32 atomics
// into out[b,dst]. total = E*4*16
// ---------------------------------------------------------------------------
__global__ void scatter_kernel(const int* __restrict__ src,
                               const int* __restrict__ dst,
                               const float* __restrict__ norm,
                               const float* __restrict__ h,
                               float* __restrict__ out, int N, size_t total) {
  size_t t = (size_t)blockIdx.x * blockDim.x + threadIdx.x;
  if (t >= total) return;
  int c4   = (int)(t & 15);
  int b    = (int)((t >> 4) & 3);
  size_t e = t >> 6;
  int s    = src[e];
  int d    = dst[e];
  float nm = norm[e];
  float4 hv = *(const float4*)(h + ((size_t)b * N + s) * C_OUT + c4 * 4);
  float* op = out + ((size_t)b * N + d) * C_OUT + c4 * 4;
  atomicAdd(op + 0, hv.x * nm);
  atomicAdd(op + 1, hv.y * nm);
  atomicAdd(op + 2, hv.z * nm);
  atomicAdd(op + 3, hv.w * nm);
}

// ---------------------------------------------------------------------------
extern "C" void kernel_launch(void* const* d_in, const int* in_sizes, int n_in,
                              void* d_out, int out_size, void* d_ws, size_t ws_size,
                              hipStream_t stream) {
  const float* x    = (const float*)d_in[0];   // [B,N,32]
  const int*   eidx = (const int*)  d_in[1];   // [2,E]
  const float* w    = (const float*)d_in[2];   // [E]
  const float* W    = (const float*)d_in[3];   // [32,64]
  const float* bias = (const float*)d_in[4];   // [64]

  const int E    = in_sizes[2];
  const int N    = in_sizes[0] / (BATCH * C_IN);
  const int rows = BATCH * N;                  // 200000, multiple of 16

  // Workspace layout (floats): dinv[N] | norm[E] | h[rows*64] (~55 MB total)
  float* ws   = (float*)d_ws;
  float* dinv = ws;                            // holds deg, then rsqrt in place
  float* norm = ws + N;
  size_t hoff = (((size_t)N + (size_t)E) + 15) & ~(size_t)15;
  float* h    = ws + hoff;
  float* out  = (float*)d_out;

  const int T = 256;
  deg_init_kernel   <<<(N + T - 1) / T, T, 0, stream>>>(dinv, N);
  deg_accum_kernel  <<<(E + T - 1) / T, T, 0, stream>>>(eidx + E, w, dinv, E);
  deg_to_dinv_kernel<<<(N + T - 1) / T, T, 0, stream>>>(dinv, N);
  edge_norm_kernel  <<<(E + T - 1) / T, T, 0, stream>>>(eidx, eidx + E, w, dinv, norm, E);

  const int tiles = (rows + 15) / 16;
  gemm_wmma_kernel<<<(tiles + 7) / 8, T, 0, stream>>>(x, W, h, rows);

  const size_t total4 = (size_t)rows * (C_OUT / 4);
  out_init_kernel<<<(unsigned)((total4 + T - 1) / T), T, 0, stream>>>(h, dinv, bias, out, N, total4);

  const size_t totalS = (size_t)E * BATCH * (C_OUT / 4);
  scatter_kernel<<<(unsigned)((totalS + T - 1) / T), T, 0, stream>>>(eidx, eidx + E, norm, h, out, N, totalS);
}